// PELoss_5497558139563
// MI455X (gfx1250) — compile-verified
//
#include <hip/hip_runtime.h>
#include <hip/hip_bf16.h>

#define ALPHA_F 2.0f
#define EPS_F 1e-8f
#define NUM_CLASSES 100
#define B_N 8192
#define D_DIM 64

typedef __attribute__((ext_vector_type(2))) float v2f;
typedef __attribute__((ext_vector_type(8))) float v8f;

__global__ void init_kernel(float* out, float* counts) {
    int t = threadIdx.x;
    if (t < NUM_CLASSES) counts[t] = 0.0f;
    if (t == 0) out[0] = 0.0f;
}

// Per-row squared norms + per-class counts (float atomics; exact for <2^24).
__global__ void prep_kernel(const float* __restrict__ emb,
                            const int* __restrict__ labels,
                            float* counts, float* sqn) {
    int i = blockIdx.x * blockDim.x + threadIdx.x;
    if (i >= B_N) return;
    const float4* row = (const float4*)(emb + (size_t)i * D_DIM);
    float s = 0.0f;
#pragma unroll
    for (int k = 0; k < D_DIM / 4; ++k) {
        float4 v = row[k];
        s += v.x * v.x + v.y * v.y + v.z * v.z + v.w * v.w;
    }
    sqn[i] = s;
    atomicAdd(&counts[labels[i]], 1.0f);
}

// PE loss: each wave owns one 16-row tile of E (A fragments resident in
// 32 VGPRs) and sweeps 32 column tiles of 16 via V_WMMA_F32_16X16X4_F32,
// masking + reducing the 16x16 Gram tile in the C/D register layout.
__global__ __launch_bounds__(128) void pe_kernel(
    const float* __restrict__ emb, const int* __restrict__ labels,
    const float* __restrict__ counts, const float* __restrict__ sqn,
    float* out) {
    const unsigned lane = threadIdx.x & 31u;
    const unsigned wave = threadIdx.x >> 5;
    const unsigned task = blockIdx.x * 4u + wave;   // 8192 wave-tasks
    const unsigned rowTile  = task >> 4;            // 512 row tiles
    const unsigned colChunk = task & 15u;           // 16 chunks x 32 col tiles

    const unsigned mrow  = lane & 15u;              // M (A) / N (B) index
    const unsigned khalf = (lane >> 4) * 2u;        // K sub-offset: 0 or 2
    const unsigned rowBase = rowTile * 16u;

    // A fragments for all 16 K-steps (K=64 total, 4 per WMMA).
    v2f a[16];
#pragma unroll
    for (int ks = 0; ks < 16; ++ks) {
        const float* p = emb + (size_t)(rowBase + mrow) * D_DIM + ks * 4 + khalf;
        a[ks] = *(const v2f*)p;
    }

    // Row-side metadata matching the C/D layout: c[v] -> M = v + 8*(lane>=16).
    const unsigned mOff = (lane >> 4) * 8u;
    int   rlab[8];
    float rsqn[8], rw[8];
#pragma unroll
    for (int v = 0; v < 8; ++v) {
        unsigned r = rowBase + mOff + v;
        rlab[v] = labels[r];
        rsqn[v] = sqn[r];
        rw[v]   = 1.0f / fmaxf(counts[rlab[v]], 1.0f);
    }

    float acc = 0.0f;
    for (unsigned t = 0; t < 32; ++t) {
        const unsigned colBase = (colChunk * 32u + t) * 16u;
        v8f c = {0.f, 0.f, 0.f, 0.f, 0.f, 0.f, 0.f, 0.f};
#pragma unroll
        for (int ks = 0; ks < 16; ++ks) {
            const float* p = emb + (size_t)(colBase + mrow) * D_DIM + ks * 4 + khalf;
            v2f b = *(const v2f*)p;
            // (neg_a, A, neg_b, B, c_mod, C, reuse_a, reuse_b)
            c = __builtin_amdgcn_wmma_f32_16x16x4_f32(
                false, a[ks], false, b, (short)0, c, false, false);
        }
        // Epilogue: mask + reduce this 16x16 Gram tile (branchless, v_rsq_f32).
        const unsigned colIdx = colBase + mrow;      // N = lane&15
        const int   clab = labels[colIdx];
        const float csqn = sqn[colIdx];
#pragma unroll
        for (int v = 0; v < 8; ++v) {
            unsigned rowIdx = rowBase + mOff + v;
            float sq = fmaxf(rsqn[v] + csqn - 2.0f * c[v], 0.0f);
            float contrib = (__builtin_amdgcn_rsqf(sq) + EPS_F) * rw[v];
            bool pair = (rlab[v] == clab) && (rowIdx != colIdx);
            acc += pair ? contrib : 0.0f;
        }
    }
    // wave32 reduction
#pragma unroll
    for (int off = 16; off > 0; off >>= 1)
        acc += __shfl_xor(acc, off, 32);
    if (lane == 0) atomicAdd(out, ALPHA_F * acc);
}

// CE loss: one wave per row of preds[8192,100].
__global__ __launch_bounds__(128) void ce_kernel(
    const float* __restrict__ preds, const int* __restrict__ labels,
    float* out) {
    const unsigned lane = threadIdx.x & 31u;
    const unsigned wave = threadIdx.x >> 5;
    const unsigned row = blockIdx.x * 4u + wave;
    if (row >= B_N) return;
    const float* p = preds + (size_t)row * NUM_CLASSES;

    float m = -INFINITY;
    for (unsigned c = lane; c < NUM_CLASSES; c += 32) m = fmaxf(m, p[c]);
#pragma unroll
    for (int off = 16; off > 0; off >>= 1)
        m = fmaxf(m, __shfl_xor(m, off, 32));

    float s = 0.0f;
    for (unsigned c = lane; c < NUM_CLASSES; c += 32) s += __expf(p[c] - m);
#pragma unroll
    for (int off = 16; off > 0; off >>= 1)
        s += __shfl_xor(s, off, 32);

    if (lane == 0) {
        float lse = m + __logf(s);
        float li = -(p[labels[row]] - lse);
        atomicAdd(out, li * (1.0f / (float)B_N));
    }
}

extern "C" void kernel_launch(void* const* d_in, const int* in_sizes, int n_in,
                              void* d_out, int out_size, void* d_ws, size_t ws_size,
                              hipStream_t stream) {
    const float* emb    = (const float*)d_in[0];   // [8192, 64] f32
    const float* preds  = (const float*)d_in[1];   // [8192, 100] f32
    const int*   labels = (const int*)d_in[2];     // [8192] int
    float* out = (float*)d_out;                    // scalar f32

    float* counts = (float*)d_ws;                  // 100 floats
    float* sqn    = (float*)d_ws + 128;            // 8192 floats

    init_kernel<<<1, 128, 0, stream>>>(out, counts);
    prep_kernel<<<B_N / 256, 256, 0, stream>>>(emb, labels, counts, sqn);
    pe_kernel<<<2048, 128, 0, stream>>>(emb, labels, counts, sqn, out);
    ce_kernel<<<2048, 128, 0, stream>>>(preds, labels, out);
}